// GAT_34351148434376
// MI455X (gfx1250) — compile-verified
//
#include <hip/hip_runtime.h>
#include <cmath>

typedef float v2f __attribute__((ext_vector_type(2)));
typedef float v8f __attribute__((ext_vector_type(8)));

#define SLOPE 0.2f

// ---------------------------------------------------------------------------
// GEMM: XW[rows,128] = X[rows,128] @ W[128,128], fp32 WMMA (16x16x4).
// One block = 256 threads = 8 waves; block owns 16 rows, wave w owns cols
// [16w, 16w+16). A tile staged in LDS (padded to 132 to avoid bank conflicts).
// ---------------------------------------------------------------------------
__global__ __launch_bounds__(256)
void gat_gemm128(const float* __restrict__ X, const float* __restrict__ W,
                 float* __restrict__ XW, int rows) {
  __shared__ float As[16][132];
  const int m0 = blockIdx.x * 16;
  for (int i = threadIdx.x; i < 16 * 128; i += 256) {
    int r = i >> 7, c = i & 127;
    int gr = m0 + r;
    As[r][c] = (gr < rows) ? X[(size_t)gr * 128 + c] : 0.0f;
  }
  __syncthreads();

  const int lane = threadIdx.x & 31;
  const int wv   = threadIdx.x >> 5;     // 0..7
  const int n0   = wv * 16;
  const int l15  = lane & 15;
  const int koff = (lane >> 4) * 2;      // lanes 16-31 hold K=2,3 halves

  v8f acc = {};
  #pragma unroll 4
  for (int k0 = 0; k0 < 128; k0 += 4) {
    v2f a, b;
    a.x = As[l15][k0 + koff];
    a.y = As[l15][k0 + koff + 1];
    b.x = W[(size_t)(k0 + koff)     * 128 + n0 + l15];
    b.y = W[(size_t)(k0 + koff + 1) * 128 + n0 + l15];
    acc = __builtin_amdgcn_wmma_f32_16x16x4_f32(false, a, false, b,
                                                (short)0, acc, false, false);
  }

  const int col   = n0 + l15;
  const int rbase = (lane < 16) ? 0 : 8;   // D: VGPR r -> M=r / M=r+8
  if (m0 + 16 <= rows) {
    // fast path: whole tile in range (always true when rows % 16 == 0)
    float* p = XW + (size_t)(m0 + rbase) * 128 + col;
    #pragma unroll
    for (int r = 0; r < 8; ++r) p[(size_t)r * 128] = acc[r];
  } else {
    #pragma unroll
    for (int r = 0; r < 8; ++r) {
      int row = m0 + rbase + r;
      if (row < rows) XW[(size_t)row * 128 + col] = acc[r];
    }
  }
}

// ---------------------------------------------------------------------------
// Per-(node,head) attention scores: s = sum_c xw[n,h,c] * a[h,c]
// ---------------------------------------------------------------------------
__global__ void gat_scores(const float* __restrict__ xw,
                           const float* __restrict__ asrc,
                           const float* __restrict__ adst,
                           float* __restrict__ ss, float* __restrict__ sd, int N_) {
  int i = blockIdx.x * blockDim.x + threadIdx.x;
  if (i >= N_ * 4) return;
  int n = i >> 2, h = i & 3;
  const float* p  = xw + (size_t)n * 128 + h * 32;
  const float* pa = asrc + h * 32;
  const float* pb = adst + h * 32;
  float s0 = 0.f, s1 = 0.f;
  #pragma unroll
  for (int c = 0; c < 32; ++c) { float v = p[c]; s0 += v * pa[c]; s1 += v * pb[c]; }
  ss[i] = s0; sd[i] = s1;
}

// ---------------------------------------------------------------------------
// Init: out accumulator = 0, amax = -inf, denom = 0
// ---------------------------------------------------------------------------
__global__ void gat_init(float* __restrict__ out, float* __restrict__ amax,
                         float* __restrict__ dnm, int nOut, int nHead) {
  int i = blockIdx.x * blockDim.x + threadIdx.x;
  if (i < nOut)  out[i] = 0.0f;
  if (i < nHead) { amax[i] = -__builtin_inff(); dnm[i] = 0.0f; }
}

__device__ __forceinline__ void atomicMaxF(float* addr, float v) {
  // sign-aware int reinterpretation: valid for mixed-sign IEEE floats
  if (v >= 0.0f) atomicMax((int*)addr, __float_as_int(v));
  else           atomicMin((unsigned int*)addr, __float_as_uint(v));
}

__device__ __forceinline__ void edge_endpoints(const int* ei, int e, int E_,
                                               int& s, int& d) {
  if (e < E_) { s = ei[e]; d = ei[E_ + e]; }   // edge_index[0][e], [1][e]
  else        { s = e - E_; d = s; }           // appended self loops
}

// ---------------------------------------------------------------------------
// Pass 1 over edges: segment max of leaky-relu scores (per head)
// ---------------------------------------------------------------------------
__global__ void gat_edge_max(const int* __restrict__ ei,
                             const float* __restrict__ ss,
                             const float* __restrict__ sd,
                             float* __restrict__ amax, int E_, int Etot) {
  int e = blockIdx.x * blockDim.x + threadIdx.x;
  if (e >= Etot) return;
  int s, d; edge_endpoints(ei, e, E_, s, d);
  #pragma unroll
  for (int h = 0; h < 4; ++h) {
    float a = ss[s * 4 + h] + sd[d * 4 + h];
    a = (a > 0.f) ? a : a * SLOPE;
    atomicMaxF(&amax[d * 4 + h], a);
  }
}

// ---------------------------------------------------------------------------
// Pass 2 over edges: denom = segment_sum(exp(alpha - amax[dst]))
// ---------------------------------------------------------------------------
__global__ void gat_edge_sum(const int* __restrict__ ei,
                             const float* __restrict__ ss,
                             const float* __restrict__ sd,
                             const float* __restrict__ amax,
                             float* __restrict__ dnm, int E_, int Etot) {
  int e = blockIdx.x * blockDim.x + threadIdx.x;
  if (e >= Etot) return;
  int s, d; edge_endpoints(ei, e, E_, s, d);
  #pragma unroll
  for (int h = 0; h < 4; ++h) {
    float a = ss[s * 4 + h] + sd[d * 4 + h];
    a = (a > 0.f) ? a : a * SLOPE;
    float mx = amax[d * 4 + h];
    if (!isfinite(mx)) mx = 0.0f;
    atomicAdd(&dnm[d * 4 + h], expf(a - mx));
  }
}

// ---------------------------------------------------------------------------
// Pass 3 over edges: wave per edge. Lane owns 4 channels (head = lane/8):
// out[dst] += xw[src] * attn   (float4 gather + 4 f32 atomic adds per lane)
// ---------------------------------------------------------------------------
__global__ void gat_edge_scatter(const int* __restrict__ ei,
                                 const float* __restrict__ ss,
                                 const float* __restrict__ sd,
                                 const float* __restrict__ amax,
                                 const float* __restrict__ dnm,
                                 const float* __restrict__ xw,
                                 float* __restrict__ out, int E_, int Etot) {
  int gid  = blockIdx.x * blockDim.x + threadIdx.x;
  int e    = gid >> 5;
  int lane = gid & 31;
  if (e >= Etot) return;
  int s, d; edge_endpoints(ei, e, E_, s, d);

  int h = lane >> 3;                       // channels [lane*4, lane*4+4) in head h
  float a = ss[s * 4 + h] + sd[d * 4 + h];
  a = (a > 0.f) ? a : a * SLOPE;
  float mx = amax[d * 4 + h];
  if (!isfinite(mx)) mx = 0.0f;
  float ex   = expf(a - mx);
  float attn = ex / fmaxf(dnm[d * 4 + h], 1e-16f);

  int c = lane * 4;
  const float4 xv = *(const float4*)(xw + (size_t)s * 128 + c);
  float* op = out + (size_t)d * 128 + c;
  atomicAdd(op + 0, xv.x * attn);
  atomicAdd(op + 1, xv.y * attn);
  atomicAdd(op + 2, xv.z * attn);
  atomicAdd(op + 3, xv.w * attn);
}

// ---------------------------------------------------------------------------
// ELU + bias (in place), and final head-mean + bias
// ---------------------------------------------------------------------------
__global__ void gat_elu_bias(float* __restrict__ h, const float* __restrict__ b, int n) {
  int i = blockIdx.x * blockDim.x + threadIdx.x;
  if (i >= n) return;
  float v = h[i] + b[i & 127];
  h[i] = (v > 0.f) ? v : expm1f(v);
}

__global__ void gat_final(const float* __restrict__ acc, const float* __restrict__ b,
                          float* __restrict__ out, int N_) {
  int i = blockIdx.x * blockDim.x + threadIdx.x;
  if (i >= N_ * 32) return;
  int n = i >> 5, c = i & 31;
  const float* p = acc + (size_t)n * 128 + c;
  out[i] = 0.25f * (p[0] + p[32] + p[64] + p[96]) + b[c];
}

// ---------------------------------------------------------------------------
extern "C" void kernel_launch(void* const* d_in, const int* in_sizes, int n_in,
                              void* d_out, int out_size, void* d_ws, size_t ws_size,
                              hipStream_t stream) {
  const float* x   = (const float*)d_in[0];
  const int*   ei  = (const int*)  d_in[1];
  const float* W1  = (const float*)d_in[2];
  const float* as1 = (const float*)d_in[3];
  const float* ad1 = (const float*)d_in[4];
  const float* b1  = (const float*)d_in[5];
  const float* W2  = (const float*)d_in[6];
  const float* as2 = (const float*)d_in[7];
  const float* ad2 = (const float*)d_in[8];
  const float* b2  = (const float*)d_in[9];
  const float* W3  = (const float*)d_in[10];
  const float* as3 = (const float*)d_in[11];
  const float* ad3 = (const float*)d_in[12];
  const float* b3  = (const float*)d_in[13];

  const int N_   = in_sizes[0] / 128;
  const int E_   = in_sizes[1] / 2;
  const int Etot = E_ + N_;

  float* A  = (float*)d_ws;              // xw        [N,128]
  float* B  = A  + (size_t)N_ * 128;     // accum / h [N,128]
  float* ss = B  + (size_t)N_ * 128;     // [N,4]
  float* sd = ss + (size_t)N_ * 4;       // [N,4]
  float* am = sd + (size_t)N_ * 4;       // [N,4]
  float* dn = am + (size_t)N_ * 4;       // [N,4]

  const int TB = 256;
  auto layer = [&](const float* hin, const float* W, const float* asx, const float* adx) {
    gat_gemm128<<<(N_ + 15) / 16, TB, 0, stream>>>(hin, W, A, N_);
    gat_scores<<<(N_ * 4 + TB - 1) / TB, TB, 0, stream>>>(A, asx, adx, ss, sd, N_);
    gat_init<<<(N_ * 128 + TB - 1) / TB, TB, 0, stream>>>(B, am, dn, N_ * 128, N_ * 4);
    gat_edge_max<<<(Etot + TB - 1) / TB, TB, 0, stream>>>(ei, ss, sd, am, E_, Etot);
    gat_edge_sum<<<(Etot + TB - 1) / TB, TB, 0, stream>>>(ei, ss, sd, am, dn, E_, Etot);
    long long thr = (long long)Etot * 32;
    gat_edge_scatter<<<(int)((thr + TB - 1) / TB), TB, 0, stream>>>(
        ei, ss, sd, am, dn, A, B, E_, Etot);
  };

  // Layer 1: x -> B (h1)
  layer(x, W1, as1, ad1);
  gat_elu_bias<<<(N_ * 128 + TB - 1) / TB, TB, 0, stream>>>(B, b1, N_ * 128);
  // Layer 2: B -> B (GEMM reads B before init re-zeros it; stream ordered)
  layer(B, W2, as2, ad2);
  gat_elu_bias<<<(N_ * 128 + TB - 1) / TB, TB, 0, stream>>>(B, b2, N_ * 128);
  // Layer 3: B -> d_out (mean over heads + b3)
  layer(B, W3, as3, ad3);
  gat_final<<<(N_ * 32 + TB - 1) / TB, TB, 0, stream>>>(B, b3, (float*)d_out, N_);
}